// CifPredictorV2_49873160241398
// MI455X (gfx1250) — compile-verified
//
#include <hip/hip_runtime.h>
#include <hip/hip_bf16.h>

// CIF predictor for MI455X (gfx1250).
// B=16, T=1024 (Text=1025 after tail pad), D=512, fp32 throughout.
// Out layout (flat f32): acoustic_embeds[16*1025*512] | token_num[16] |
//                        alphas[16*1025] | cif_peak[16*1025]

#define B_   16
#define T_   1024
#define TE_  1025
#define D_   512

typedef float v2f __attribute__((ext_vector_type(2)));
typedef float v8f __attribute__((ext_vector_type(8)));

// ---------------------------------------------------------------------------
// Kernel 1: alphas[b,t] = relu(sigmoid(hidden[b,t,:]·w + bias)) , t < 1024.
// One wave per row; float4 coalesced loads; shuffle reduction (wave32).
// ---------------------------------------------------------------------------
__global__ void cif_alpha_gemv(const float* __restrict__ hidden,
                               const float* __restrict__ w,
                               const float* __restrict__ bias,
                               float* __restrict__ outAlpha) {
    const int gwave = (blockIdx.x * blockDim.x + threadIdx.x) >> 5;
    const int lane  = threadIdx.x & 31;
    if (gwave >= B_ * T_) return;

    const float4* r4 = (const float4*)(hidden + (size_t)gwave * D_);
    const float4* w4 = (const float4*)w;
    float acc = 0.f;
#pragma unroll
    for (int j = 0; j < 4; ++j) {          // 32 lanes * 4 float4 = 512 floats
        float4 h  = r4[lane + 32 * j];
        float4 ww = w4[lane + 32 * j];
        acc += h.x * ww.x + h.y * ww.y + h.z * ww.z + h.w * ww.w;
    }
#pragma unroll
    for (int off = 16; off > 0; off >>= 1) acc += __shfl_xor(acc, off, 32);

    if (lane == 0) {
        float a = 1.f / (1.f + expf(-(acc + bias[0])));   // sigmoid
        a = fmaxf(a * 1.0f - 0.0f, 0.f);                  // smooth/noise/relu
        const int b = gwave / T_, t = gwave % T_;
        outAlpha[b * TE_ + t] = a;
    }
}

// ---------------------------------------------------------------------------
// Kernel 2: per-batch sequential integrate-and-fire scan (exact reference
// FP order). One wave per batch; alphas loaded 32-wide and broadcast via
// __shfl so the serial chain is pure VALU. Emits per (b,t) a packed 16-byte
// meta record {wMain, wSpill, seg, 0}:
//   wMain = cur (alpha, or dist_completion at a fire), targets embed `seg`
//   wSpill= remainds at a fire (targets embed seg+1), else 0
// plus cif_peak, token_num, fireTime[k] (t of k-th fire), nFires.
// ---------------------------------------------------------------------------
__global__ void cif_scan(float*  __restrict__ outAlpha,   // [B_, TE_] (writes tail)
                         float*  __restrict__ outPeak,    // [B_, TE_]
                         float*  __restrict__ outTok,     // [B_]
                         float4* __restrict__ meta,       // [B_, TE_] packed
                         int*    __restrict__ fireTime,
                         int*    __restrict__ nFires) {
    const int b    = blockIdx.x;
    const int lane = threadIdx.x;

    float integrate = 0.f, S = 0.f;
    int cnt = 0;

    const int nChunk = (TE_ + 31) / 32;
    for (int c = 0; c < nChunk; ++c) {
        const int t = c * 32 + lane;
        float aL = 0.f;                       // tail step t==1024 -> alpha 0
        if (t < T_) aL = outAlpha[b * TE_ + t];

        float myPeak = 0.f, myWM = 0.f, myWS = 0.f;
        int   mySeg  = 0;

        const int lim = min(32, TE_ - c * 32);
        for (int j = 0; j < lim; ++j) {
            const float a    = __shfl(aL, j, 32);
            const float dist = 1.f - integrate;
            integrate += a;
            S         += a;
            const bool  fire = (integrate >= 1.0f);
            const float cur  = fire ? dist : a;
            const float rem  = a - cur;
            if (lane == j) {
                myPeak = integrate;           // pre-subtract value
                myWM   = cur;
                myWS   = fire ? rem : 0.f;
                mySeg  = cnt;                 // fire: own embed; else: next embed
                if (fire) fireTime[b * TE_ + cnt] = c * 32 + j;
            }
            if (fire) { ++cnt; integrate -= 1.f; }
        }
        if (t < TE_) {
            outPeak[b * TE_ + t] = myPeak;
            float4 m;
            m.x = myWM;
            m.y = myWS;
            m.z = __int_as_float(mySeg);
            m.w = 0.f;
            meta[b * TE_ + t] = m;            // single b128 store
            if (t == T_) outAlpha[b * TE_ + t] = 0.f;   // tail alpha output
        }
    }
    if (lane == 0) { nFires[b] = cnt; outTok[b] = floorf(S); }
}

// ---------------------------------------------------------------------------
// Kernel 3: packed embeds via banded sparse GEMM on V_WMMA_F32_16X16X4_F32.
// One wave -> 16 embeds x 16 channels; K walks the tile's covering t-range in
// chunks of 4. A-fragment (16x4 f32): lane L holds row M=L%16, K={0,1}
// (lanes 0-15) / {2,3} (lanes 16-31). B-fragment (4x16 f32): lane L holds
// col N=L%16, same K split. C/D: VGPR j = rows j / j+8.
// Main loop is fully branchless (unconditional b128 meta + b32 hidden loads,
// v_cndmask selects); one guarded tail chunk handles the range end.
// ---------------------------------------------------------------------------
__global__ void cif_pack_wmma(const float*  __restrict__ hidden,
                              const float4* __restrict__ meta,
                              const int*    __restrict__ fireTime,
                              const int*    __restrict__ nFires,
                              float*        __restrict__ outEmb) {
    const int b    = blockIdx.z;
    const int k0   = blockIdx.x * 16;     // embed tile base
    const int d0   = blockIdx.y * 16;     // channel tile base
    const int lane = threadIdx.x;
    const int row  = lane & 15;
    const int half = lane >> 4;

    const int nf = nFires[b];
    v8f acc = {};                          // zero C accumulator

    if (k0 < nf) {                         // uniform per wave
        const int tstart = (k0 == 0) ? 0 : fireTime[b * TE_ + k0 - 1];
        const int klast  = min(k0 + 15, nf - 1);
        const int tend   = fireTime[b * TE_ + klast];  // <= 1023 (t=1024 never fires)
        const int mA     = k0 + row;       // this lane's A row (embed index)
        const int kk0    = half * 2;       // this lane's first K slot

        // Lane-steady pointers, constant stride per K-chunk.
        const float4* mp = meta + b * TE_ + tstart + kk0;             // +4 /iter
        const float*  hp = hidden + ((size_t)b * T_ + tstart + kk0) * D_
                                  + d0 + row;                         // +4*D_ /iter

        int tc = tstart;
        // ---- main loop: tc+3 <= tend, no clamps, no range selects ----
        for (; tc + 3 <= tend; tc += 4, mp += 4, hp += 4 * D_) {
            const float4 m0 = mp[0];
            const float4 m1 = mp[1];
            v2f bfrag;
            bfrag.x = hp[0];
            bfrag.y = hp[D_];

            const int s0 = __float_as_int(m0.z);
            const int s1 = __float_as_int(m1.z);
            v2f afrag;
            afrag.x = ((s0 == mA) ? m0.x : 0.f) + ((s0 + 1 == mA) ? m0.y : 0.f);
            afrag.y = ((s1 == mA) ? m1.x : 0.f) + ((s1 + 1 == mA) ? m1.y : 0.f);

            acc = __builtin_amdgcn_wmma_f32_16x16x4_f32(
                false, afrag, false, bfrag, (short)0, acc, false, false);
        }
        // ---- single guarded tail chunk (covers remaining tc..tend) ----
        if (tc <= tend) {
            const int t0 = tc + kk0;
            const int t1 = t0 + 1;
            const int tm0 = min(t0, TE_ - 1);
            const int tm1 = min(t1, TE_ - 1);
            const float4 m0 = meta[b * TE_ + tm0];
            const float4 m1 = meta[b * TE_ + tm1];
            v2f bfrag;
            bfrag.x = hidden[((size_t)b * T_ + min(t0, T_ - 1)) * D_ + d0 + row];
            bfrag.y = hidden[((size_t)b * T_ + min(t1, T_ - 1)) * D_ + d0 + row];

            const int s0 = __float_as_int(m0.z);
            const int s1 = __float_as_int(m1.z);
            float ax = ((s0 == mA) ? m0.x : 0.f) + ((s0 + 1 == mA) ? m0.y : 0.f);
            float ay = ((s1 == mA) ? m1.x : 0.f) + ((s1 + 1 == mA) ? m1.y : 0.f);
            v2f afrag;
            afrag.x = (t0 <= tend) ? ax : 0.f;
            afrag.y = (t1 <= tend) ? ay : 0.f;

            acc = __builtin_amdgcn_wmma_f32_16x16x4_f32(
                false, afrag, false, bfrag, (short)0, acc, false, false);
        }
    }

    // Store D tile; embeds >= nFires (and tile overhang) are zero.
#pragma unroll
    for (int j = 0; j < 8; ++j) {
        const int m = k0 + j + half * 8;
        if (m < TE_) {
            const float v = (m < nf) ? acc[j] : 0.f;
            outEmb[((size_t)b * TE_ + m) * D_ + d0 + row] = v;
        }
    }
}

// ---------------------------------------------------------------------------
extern "C" void kernel_launch(void* const* d_in, const int* in_sizes, int n_in,
                              void* d_out, int out_size, void* d_ws, size_t ws_size,
                              hipStream_t stream) {
    const float* hidden = (const float*)d_in[0];   // [16,1024,512]
    const float* w_cif  = (const float*)d_in[1];   // [512]
    const float* b_cif  = (const float*)d_in[2];   // [1]

    float* out      = (float*)d_out;
    float* outEmb   = out;                                   // [16,1025,512]
    float* outTok   = outEmb + (size_t)B_ * TE_ * D_;        // [16]
    float* outAlpha = outTok + B_;                           // [16,1025]
    float* outPeak  = outAlpha + (size_t)B_ * TE_;           // [16,1025]

    // Workspace: meta float4[16*1025] (262.4 KB) + fireTime + nFires (~66 KB)
    float4* meta     = (float4*)d_ws;
    int*    fireTime = (int*)(meta + (size_t)B_ * TE_);
    int*    nFires   = fireTime + (size_t)B_ * TE_;

    // K1: 16384 rows, 8 waves (256 thr) per block -> 2048 blocks.
    cif_alpha_gemv<<<dim3((B_ * T_) / 8), dim3(256), 0, stream>>>(
        hidden, w_cif, b_cif, outAlpha);

    // K2: one wave per batch.
    cif_scan<<<dim3(B_), dim3(32), 0, stream>>>(
        outAlpha, outPeak, outTok, meta, fireTime, nFires);

    // K3: (65 embed tiles) x (32 channel tiles) x (16 batches), 1 wave each.
    cif_pack_wmma<<<dim3((TE_ + 15) / 16, D_ / 16, B_), dim3(32), 0, stream>>>(
        hidden, meta, fireTime, nFires, outEmb);
}